// LIF_TBN_27848567947917
// MI455X (gfx1250) — compile-verified
//
#include <hip/hip_runtime.h>
#include <cstdint>
#include <cstddef>

// LIF recurrence, time-major (T, B, N) = (16, 128, 16384), fp32.
// Memory-bound streaming kernel: 256 MiB total traffic, ~11.5 us at 23.3 TB/s.
// CDNA5 path used: GLOBAL_LOAD_ASYNC_TO_LDS_B128 (ASYNCcnt) double-buffered
// through LDS, ds_load_b128 consumption, non-temporal global_store_b128.

typedef float v4f __attribute__((ext_vector_type(4)));

#define LIF_T   16
#define TAUF    0.5f
#define VTHF    1.0f
#define BLOCK   256
#define VEC     4
#define TILE_ELEMS (BLOCK * VEC)        // 1024 floats per buffer per block
#define TILE_BYTES (TILE_ELEMS * 4)     // 4096 bytes

__global__ __launch_bounds__(BLOCK)
void lif_tbn_async_kernel(const float* __restrict__ x,
                          float* __restrict__ y,
                          unsigned int BN) {
    __shared__ __align__(16) float buf[2][TILE_ELEMS];

    const unsigned int tid  = threadIdx.x;
    const unsigned int base = blockIdx.x * TILE_ELEMS + tid * VEC;
    if (base >= BN) return;   // exact division for the reference dims; safety only

    // Generic pointers to LDS carry the LDS byte offset in their low 32 bits
    // (aperture lives in the high dword) -> this is exactly what the async
    // instruction's VDST (LDS address VGPR) wants.
    const uint32_t lds0 = (uint32_t)(uintptr_t)(&buf[0][0]) + tid * (VEC * 4u);
    const uint32_t lds1 = lds0 + TILE_BYTES;

    const float* xp = x + base;
    float*       yp = y + base;

    // ---- Prologue: prefetch t = 0 and t = 1 (depth-2 async pipeline) ----
    {
        uint64_t g0 = (uint64_t)(uintptr_t)(xp);
        asm volatile("global_load_async_to_lds_b128 %0, %1, off"
                     :: "v"(lds0), "v"(g0) : "memory");
        uint64_t g1 = (uint64_t)(uintptr_t)(xp + (size_t)BN);
        asm volatile("global_load_async_to_lds_b128 %0, %1, off"
                     :: "v"(lds1), "v"(g1) : "memory");
    }

    float v0 = 0.0f, v1 = 0.0f, v2 = 0.0f, v3 = 0.0f;

    #pragma unroll
    for (int t = 0; t < LIF_T; ++t) {
        // Async loads complete in order: with loads 0..t+1 issued, allowing
        // one outstanding guarantees load t has landed in LDS.
        if (t + 1 < LIF_T) {
            asm volatile("s_wait_asynccnt 1" ::: "memory");
        } else {
            asm volatile("s_wait_asynccnt 0" ::: "memory");
        }

        const v4f xv = *reinterpret_cast<const v4f*>(&buf[t & 1][tid * VEC]);

        // Async-unit LDS writes are unordered vs. our ds_load: drain DScnt
        // before re-issuing a load that overwrites this buffer (t+2 hits t&1).
        asm volatile("s_wait_dscnt 0" ::: "memory");

        if (t + 2 < LIF_T) {
            uint64_t g  = (uint64_t)(uintptr_t)(xp + (size_t)(t + 2) * BN);
            uint32_t dst = ((t + 2) & 1) ? lds1 : lds0;
            asm volatile("global_load_async_to_lds_b128 %0, %1, off"
                         :: "v"(dst), "v"(g) : "memory");
        }

        // LIF step: m = 0.5*v + x ; spike = (m >= 1) ; v = spike ? 0 : m.
        // fmaf(v, 0.5f, x) is bit-exact vs mul+add since *0.5 is exact.
        v4f out;
        { float m = fmaf(v0, TAUF, xv.x); bool s = (m >= VTHF); out.x = s ? 1.0f : 0.0f; v0 = s ? 0.0f : m; }
        { float m = fmaf(v1, TAUF, xv.y); bool s = (m >= VTHF); out.y = s ? 1.0f : 0.0f; v1 = s ? 0.0f : m; }
        { float m = fmaf(v2, TAUF, xv.z); bool s = (m >= VTHF); out.z = s ? 1.0f : 0.0f; v2 = s ? 0.0f : m; }
        { float m = fmaf(v3, TAUF, xv.w); bool s = (m >= VTHF); out.w = s ? 1.0f : 0.0f; v3 = s ? 0.0f : m; }

        // Streamed once -> non-temporal b128 store (don't pollute L2).
        __builtin_nontemporal_store(out, reinterpret_cast<v4f*>(yp + (size_t)t * BN));
    }
}

extern "C" void kernel_launch(void* const* d_in, const int* in_sizes, int n_in,
                              void* d_out, int out_size, void* d_ws, size_t ws_size,
                              hipStream_t stream) {
    (void)n_in; (void)out_size; (void)d_ws; (void)ws_size;

    const float* x = (const float*)d_in[0];
    float*       y = (float*)d_out;

    const unsigned int total = (unsigned int)in_sizes[0];   // T*B*N
    const unsigned int BN    = total / LIF_T;               // 2,097,152
    const unsigned int grid  = (BN + TILE_ELEMS - 1) / TILE_ELEMS;  // 2048

    lif_tbn_async_kernel<<<grid, BLOCK, 0, stream>>>(x, y, BN);
}